// GNNMixingLayer_23888608100424
// MI455X (gfx1250) — compile-verified
//
#include <hip/hip_runtime.h>

// GNN head-mixing layer for MI455X (gfx1250, wave32, WGP).
//
//   out[i] = relu( lambda_i * (H_i @ W_self) + (sum_j G_ij H_j) @ W_neighbor )
//
// All matrix work on V_WMMA_F32_16X16X4_F32 (fp32-exact):
//   phase 1: Zstack(16x1024) = G_gated(16x16) @ Hstack   (16 WMMA / wave / tile)
//   phase 2: out_i = [lam*H_i | Z_i](16x128) @ [Ws;Wn](128x64)  (128 WMMA / wave / tile)
// H tiles stream global->LDS via double-buffered GLOBAL_LOAD_ASYNC_TO_LDS_B128
// (ASYNCcnt); weights pre-swizzled in LDS to exact WMMA B-fragment lane order.

typedef __attribute__((ext_vector_type(2))) float v2f;
typedef __attribute__((ext_vector_type(4))) float f4;
typedef __attribute__((ext_vector_type(8))) float v8f;

#define NUM_HEADS 16
#define D_HEAD    64
#define ROWS      32768               // BATCH * SEQ
#define TILE_M    16
#define NTILES    (ROWS / TILE_M)     // 2048
#define THREADS   512                 // 16 waves: wave i = output head i
#define GRID      1024
#define TPB       (NTILES / GRID)     // 2 tiles per block (double-buffer pipeline)

// LDS strides (floats)
#define RSTR      68                  // row stride: 4m banks -> conflict-free A frags
#define HSTR      (16 * RSTR + 8)     // 1096: head stride in sH (== 8 mod 64 banks)
#define HBUF      (NUM_HEADS * HSTR)  // 17536 floats per H buffer
#define ZSTR      (16 * RSTR + 4)     // 1092: head stride in sZ (8*ZSTR == 32 mod 64)
#define WBUF      4096                // one swizzled 64x64 weight matrix
#define SMEM_FLOATS (2 * HBUF + NUM_HEADS * ZSTR + 2 * WBUF)
#define SMEM_BYTES  (SMEM_FLOATS * 4) // 242,944 B < 320KB/WGP

#define ASYNC_WAIT_ALL()  asm volatile("s_wait_asynccnt 0x0" ::: "memory")

__device__ __forceinline__ v8f wmma_f32(v2f a, v2f b, v8f c) {
    return __builtin_amdgcn_wmma_f32_16x16x4_f32(false, a, false, b, (short)0, c,
                                                 false, false);
}

// Issue 8 async 16B copies per thread: one 16-head x 16-row x 64-float tile.
__device__ __forceinline__ void stage_tile_async(const float* __restrict__ H,
                                                 float* sHbuf, int row0, int tid) {
#pragma unroll
    for (int k = 0; k < 8; ++k) {
        const int fid = tid + k * THREADS;    // 0..4095 float4s
        const int seg = fid & 15;             // float4 within a 64-float row
        const int r   = fid >> 4;             // j*16 + m
        const int j   = r >> 4;
        const int m   = r & 15;
        const float*   gp  = H + ((size_t)j * ROWS + row0 + m) * D_HEAD + seg * 4;
        const unsigned lds = (unsigned)(uintptr_t)(sHbuf + j * HSTR + m * RSTR + seg * 4);
        asm volatile("global_load_async_to_lds_b128 %0, %1, off"
                     :: "v"(lds), "v"(gp) : "memory");
    }
}

__global__ __launch_bounds__(THREADS, 1)
void gnn_mix_kernel(const float* __restrict__ H,     // [16][32768][64]
                    const float* __restrict__ adj,   // [16][16]
                    const float* __restrict__ Wself, // [64][64]
                    const float* __restrict__ Wneig, // [64][64]
                    const float* __restrict__ lam,   // [16]
                    const float* __restrict__ gij,   // [16][16]
                    float* __restrict__ out)         // [16][32768][64]
{
    extern __shared__ float smem[];
    float* sH  = smem;                      // [2][HBUF]
    float* sZ  = smem + 2 * HBUF;           // [16][ZSTR]
    float* sBF = sZ + NUM_HEADS * ZSTR;     // [2][WBUF] swizzled Ws, Wn

    const int tid  = threadIdx.x;
    const int wave = tid >> 5;
    const int lane = tid & 31;
    const int head = wave;

    // WMMA fp32 lane mapping (16x4 A / 4x16 B, 2 VGPRs each):
    //   lanes 0-15: K = k0,k0+1 ; lanes 16-31: K = k0+2,k0+3 ; M(or N) = lane&15
    const int mrow = lane & 15;
    const int kg2  = (lane >> 4) * 2;
    const int ncol = lane & 15;

    // ---- stage weights, pre-swizzled to exact B-fragment lane order ----
    // sBF[sel][((ks*4 + nt)*32 + lane)*2 + koff] = W[4ks + 2*(lane>>4) + koff][nt*16 + (lane&15)]
    for (int fid = tid; fid < 1024; fid += THREADS) {   // 1024 float4 per matrix
        const int k  = fid >> 4;
        const int c0 = (fid & 15) * 4;
        const f4 vs = ((const f4*)(Wself + k * 64))[fid & 15];
        const f4 vn = ((const f4*)(Wneig + k * 64))[fid & 15];
        const int ks = k >> 2, grp = (k >> 1) & 1, koff = k & 1;
#pragma unroll
        for (int q = 0; q < 4; ++q) {
            const int n   = c0 + q;
            const int idx = (((ks * 4 + (n >> 4)) * 32) + grp * 16 + (n & 15)) * 2 + koff;
            sBF[idx]        = vs[q];
            sBF[WBUF + idx] = vn[q];
        }
    }

    // ---- gated G as a WMMA A-fragment (rows i = lane&15, K = head j) ----
    v2f aG[4];
#pragma unroll
    for (int kk = 0; kk < 4; ++kk) {
        const int j = kk * 4 + kg2;
        const int i = lane & 15;
        const float a0 = adj[i * 16 + j],     g0 = gij[i * 16 + j];
        const float a1 = adj[i * 16 + j + 1], g1 = gij[i * 16 + j + 1];
        aG[kk].x = (a0 > 0.0f) ? g0 : 0.0f;
        aG[kk].y = (a1 > 0.0f) ? g1 : 0.0f;
    }
    const float lam_i = lam[head];

    const int t0 = blockIdx.x * TPB;
    stage_tile_async(H, sH, t0 * TILE_M, tid);          // prologue -> buf 0

    for (int it = 0; it < TPB; ++it) {
        const int  tile = t0 + it;
        const int  buf  = it & 1;
        const int  row0 = tile * TILE_M;
        float* sHb = sH + buf * HBUF;

        ASYNC_WAIT_ALL();      // my 8 async loads for this tile are in LDS
        __syncthreads();       // everyone's loads landed; prev-iter readers done

        if (it + 1 < TPB)      // prefetch next tile into the other buffer
            stage_tile_async(H, sH + (buf ^ 1) * HBUF, (tile + 1) * TILE_M, tid);

        // ---- phase 1: Z = G_gated @ Hstack via WMMA ----
        // wave w owns tile-row m = w; D rows span all 16 output heads.
#pragma unroll
        for (int db = 0; db < 4; ++db) {
            v8f zacc = {};
#pragma unroll
            for (int kk = 0; kk < 4; ++kk) {
                const int j = kk * 4 + kg2;
                v2f b;
                b.x = sHb[ j      * HSTR + wave * RSTR + db * 16 + ncol];
                b.y = sHb[(j + 1) * HSTR + wave * RSTR + db * 16 + ncol];
                zacc = wmma_f32(aG[kk], b, zacc);
            }
#pragma unroll
            for (int v = 0; v < 8; ++v) {
                const int iv = v + ((lane >> 4) << 3);   // output head index
                sZ[iv * ZSTR + wave * RSTR + db * 16 + ncol] = zacc[v];
            }
        }

        // ---- A fragments for the self term: lambda_i * H_i ----
        v2f aS[16];
        const float* hrow = sHb + head * HSTR + mrow * RSTR;
#pragma unroll
        for (int ks = 0; ks < 16; ++ks) {
            const v2f h = *(const v2f*)(hrow + ks * 4 + kg2);
            aS[ks].x = lam_i * h.x;
            aS[ks].y = lam_i * h.y;
        }

        __syncthreads();       // sZ complete across waves

        v2f aZ[16];
        const float* zrow = sZ + head * ZSTR + mrow * RSTR;
#pragma unroll
        for (int ks = 0; ks < 16; ++ks)
            aZ[ks] = *(const v2f*)(zrow + ks * 4 + kg2);

        // ---- phase 2: [lam*H_i | Z_i] @ [Ws ; Wn], fused accumulation ----
#pragma unroll
        for (int nt = 0; nt < 4; ++nt) {
            v8f acc = {};
#pragma unroll
            for (int ks = 0; ks < 16; ++ks) {
                const v2f b = *(const v2f*)(sBF + ((ks * 4 + nt) * 32 + lane) * 2);
                acc = wmma_f32(aS[ks], b, acc);
            }
#pragma unroll
            for (int ks = 0; ks < 16; ++ks) {
                const v2f b = *(const v2f*)(sBF + WBUF + ((ks * 4 + nt) * 32 + lane) * 2);
                acc = wmma_f32(aZ[ks], b, acc);
            }
            // D layout: VGPR v -> M = v + 8*(lane>>4), N = nt*16 + (lane&15)
            const int mbase = (lane >> 4) * 8;
#pragma unroll
            for (int v = 0; v < 8; ++v) {
                float r = acc[v];
                r = r > 0.0f ? r : 0.0f;
                out[((size_t)head * ROWS + row0 + mbase + v) * D_HEAD + nt * 16 + ncol] = r;
            }
        }
        // next iteration's s_wait + barrier fences sZ/sH reuse
    }
}

extern "C" void kernel_launch(void* const* d_in, const int* in_sizes, int n_in,
                              void* d_out, int out_size, void* d_ws, size_t ws_size,
                              hipStream_t stream) {
    const float* H   = (const float*)d_in[0]; // head_outputs
    const float* adj = (const float*)d_in[1]; // adjacency
    const float* Ws  = (const float*)d_in[2]; // W_self
    const float* Wn  = (const float*)d_in[3]; // W_neighbor
    const float* lm  = (const float*)d_in[4]; // lambda_self
    const float* g   = (const float*)d_in[5]; // g_ij
    float* o         = (float*)d_out;

    (void)hipFuncSetAttribute((const void*)gnn_mix_kernel,
                              hipFuncAttributeMaxDynamicSharedMemorySize,
                              SMEM_BYTES);

    hipLaunchKernelGGL(gnn_mix_kernel, dim3(GRID), dim3(THREADS), SMEM_BYTES, stream,
                       H, adj, Ws, Wn, lm, g, o);
}